// GCN_41154376630597
// MI455X (gfx1250) — compile-verified
//
#include <hip/hip_runtime.h>
#include <hip/hip_bf16.h>
#include <cstdint>
#include <cstddef>

typedef __attribute__((ext_vector_type(2))) float v2f;
typedef __attribute__((ext_vector_type(8))) float v8f;

#define BN_EPS 1e-5f

// ---------------------------------------------------------------- zero fill
__global__ void k_zero(float* __restrict__ p, size_t n) {
  size_t i = (size_t)blockIdx.x * blockDim.x + threadIdx.x;
  if (i < n) p[i] = 0.0f;
}

// ---------------------------------------------------------------- degree count
__global__ void k_deg(const int* __restrict__ src, const int* __restrict__ dst,
                      float* __restrict__ odeg, float* __restrict__ ideg, int ne) {
  int e = blockIdx.x * blockDim.x + threadIdx.x;
  if (e >= ne) return;
  atomicAdd(&odeg[src[e]], 1.0f);
  atomicAdd(&ideg[dst[e]], 1.0f);
}

// ---------------------------------------------------------------- deg -> deg^{-1/2}
__global__ void k_norm(float* __restrict__ p, int n) {
  int i = blockIdx.x * blockDim.x + threadIdx.x;
  if (i >= n) return;
  float d = p[i];
  p[i] = (d > 0.0f) ? rsqrtf(fmaxf(d, 1.0f)) : 0.0f;
}

// ---------------------------------------------------------------- GEMM via fp32 WMMA
// Y[n,DOUT] = (diag(norm) * X[n,128]) @ W[128,DOUT]
//
// W is staged transposed + zero-padded into LDS once per block:
//   sW[n * LSTR + k],  n in [0, NT*16),  LSTR = 132 dwords.
// Bank math: lane bank = (n*132 + kb) % 64 = (4*lid + 2*half) + const -> each
// wave ds_load_b64 touches all 64 banks exactly once (conflict-free).
//
// Fragments (V_WMMA_F32_16X16X4_F32, wave32):
//   A 16x4  (2 VGPRs): lane m = lane&15, K-pair {0,1}/{2,3} by lane>>4
//   B 4x16  (2 VGPRs): lane n = lane&15, K-pair by lane>>4
//   C 16x16 (v8f):     VGPR r -> row r (+8 for upper half-wave), col = lane&15
template<int NT, int DOUT>
__global__ void k_gemm_wmma(const float* __restrict__ X,
                            const float* __restrict__ norm,
                            const float* __restrict__ W,   // [128, DOUT] row-major
                            float* __restrict__ Y,         // [n, DOUT]
                            int n_tiles_m) {
  const int K = 128;
  const int NPAD = NT * 16;           // padded column count (tile multiple)
  const int LSTR = K + 4;             // 132: conflict-free LDS row stride
  __shared__ float sW[NPAD * LSTR];

  // cooperative fill: transpose W[k][n] -> sW[n][k]; zero-pad n >= DOUT
  for (int idx = threadIdx.x; idx < NPAD * K; idx += blockDim.x) {
    int k = idx & (K - 1);
    int n = idx >> 7;
    float v = (n < DOUT) ? W[(size_t)k * DOUT + n] : 0.0f;
    sW[n * LSTR + k] = v;
  }
  __syncthreads();

  int wave = blockIdx.x * (blockDim.x >> 5) + (threadIdx.x >> 5);
  if (wave >= n_tiles_m) return;      // uniform per wave
  int lane = threadIdx.x & 31;
  int half = lane >> 4;               // 0 or 1
  int lid  = lane & 15;
  int m  = wave * 16 + lid;
  int kb = half * 2;                  // K offset pair {0,1} or {2,3}
  float s = norm[m];
  const float* __restrict__ Arow = X + (size_t)m * K + kb;

  int bbase[NT];
#pragma unroll
  for (int t = 0; t < NT; ++t) bbase[t] = (t * 16 + lid) * LSTR + kb;

  v8f zero = {};
  v8f acc[NT];
#pragma unroll
  for (int t = 0; t < NT; ++t) acc[t] = zero;

#pragma unroll 4
  for (int k0 = 0; k0 < K; k0 += 4) {
    v2f a = *(const v2f*)(Arow + k0);
    a.x *= s;
    a.y *= s;
#pragma unroll
    for (int t = 0; t < NT; ++t) {
      v2f b = *(const v2f*)(&sW[bbase[t] + k0]);
      acc[t] = __builtin_amdgcn_wmma_f32_16x16x4_f32(
          false, a, false, b, (short)0, acc[t], false, false);
    }
  }

#pragma unroll
  for (int t = 0; t < NT; ++t) {
    int ncol = t * 16 + lid;
    if (ncol >= DOUT) continue;       // compile-time eliminable for DOUT=128
#pragma unroll
    for (int r = 0; r < 8; ++r) {
      int row = wave * 16 + r + half * 8;
      Y[(size_t)row * DOUT + ncol] = acc[t][r];
    }
  }
}

// ---------------------------------------------------------------- SpMM scatter-add
// agg[dst[e], :] += Xe[src[e], :]; one thread per (edge, 4-float chunk).
__global__ void k_spmm(const float* __restrict__ Xe,
                       const int* __restrict__ src, const int* __restrict__ dst,
                       float* __restrict__ agg, int ne, int dout) {
  unsigned chunks = (unsigned)(dout >> 2);
  unsigned idx = blockIdx.x * blockDim.x + threadIdx.x;
  unsigned total = (unsigned)ne * chunks;
  if (idx >= total) return;
  unsigned e = idx / chunks;
  unsigned q = idx - e * chunks;
  int sN = src[e], dN = dst[e];
  const float4 v = *(const float4*)(Xe + (size_t)sN * dout + (size_t)q * 4);
  float* p = agg + (size_t)dN * dout + (size_t)q * 4;
  atomicAdd(p + 0, v.x);
  atomicAdd(p + 1, v.y);
  atomicAdd(p + 2, v.z);
  atomicAdd(p + 3, v.w);
}

// ---------------------------------------------------------------- BN stats (sum, sumsq) per channel
// blockDim.x == d; each block strides over rows; coalesced channel reads.
__global__ void k_bn_stats(const float* __restrict__ agg, const float* __restrict__ innorm,
                           float* __restrict__ stats, int n, int d) {
  int c = threadIdx.x;
  float sum = 0.0f, sq = 0.0f;
  for (int i = blockIdx.x; i < n; i += gridDim.x) {
    float v = agg[(size_t)i * d + c] * innorm[i];
    sum += v;
    sq  += v * v;
  }
  atomicAdd(&stats[c], sum);
  atomicAdd(&stats[d + c], sq);
}

// ---------------------------------------------------------------- BN normalize + ReLU (fused with in_norm scale)
template<int D>
__global__ void k_bn_relu(const float* __restrict__ agg, const float* __restrict__ innorm,
                          const float* __restrict__ stats,
                          const float* __restrict__ gamma, const float* __restrict__ beta,
                          float* __restrict__ out, int n) {
  size_t idx = (size_t)blockIdx.x * blockDim.x + threadIdx.x;
  if (idx >= (size_t)n * D) return;
  int c = (int)(idx & (D - 1));      // D is a power of two (128)
  int i = (int)(idx >> 7);
  float invn = 1.0f / (float)n;
  float mean = stats[c] * invn;
  float var  = stats[D + c] * invn - mean * mean;
  float v = agg[idx] * innorm[i];
  v = (v - mean) * rsqrtf(var + BN_EPS) * gamma[c] + beta[c];
  out[idx] = fmaxf(v, 0.0f);
}

// ---------------------------------------------------------------- final layer: in_norm scale + bias
__global__ void k_final(const float* __restrict__ agg, const float* __restrict__ innorm,
                        const float* __restrict__ b2, float* __restrict__ out,
                        int n, int d) {
  size_t idx = (size_t)blockIdx.x * blockDim.x + threadIdx.x;
  if (idx >= (size_t)n * d) return;
  int c = (int)(idx % (unsigned)d);
  int i = (int)(idx / (unsigned)d);
  out[idx] = agg[idx] * innorm[i] + b2[c];
}

// ================================================================ launch
extern "C" void kernel_launch(void* const* d_in, const int* in_sizes, int n_in,
                              void* d_out, int out_size, void* d_ws, size_t ws_size,
                              hipStream_t stream) {
  const float* feat  = (const float*)d_in[0];
  const int*   src   = (const int*)d_in[1];
  const int*   dst   = (const int*)d_in[2];
  const float* W0    = (const float*)d_in[3];
  const float* W1    = (const float*)d_in[4];
  const float* W2    = (const float*)d_in[5];
  const float* b2    = (const float*)d_in[6];
  const float* g0    = (const float*)d_in[7];
  const float* beta0 = (const float*)d_in[8];
  const float* g1    = (const float*)d_in[9];
  const float* beta1 = (const float*)d_in[10];
  float* out = (float*)d_out;

  const int D = 128;                 // feature / hidden dim
  const int C = 40;                  // output classes
  const int N  = in_sizes[0] / D;    // 100000 (multiple of 16)
  const int NE = in_sizes[1];        // 1600000

  // workspace layout (floats)
  float* ws    = (float*)d_ws;
  float* odeg  = ws;                              // N  (becomes out_norm)
  float* ideg  = ws + N;                          // N  (becomes in_norm)
  float* stats = ws + 2 * (size_t)N;              // 256
  float* bufA  = stats + 256;                     // N*128 (gemm out)
  float* bufB  = bufA + (size_t)N * D;            // N*128 (agg)
  float* bufC  = bufB + (size_t)N * D;            // N*128 (activations)

  const int TPB = 256;
  int mtiles = N / 16;                            // 6250
  dim3 gemm_blk(128);                             // 4 waves per block
  dim3 gemm_grd((mtiles + 3) / 4);

  // --- degrees -> norms ---
  k_zero<<<(unsigned)((2 * (size_t)N + 256 + TPB - 1) / TPB), TPB, 0, stream>>>(odeg, 2 * (size_t)N + 256);
  k_deg<<<(NE + TPB - 1) / TPB, TPB, 0, stream>>>(src, dst, odeg, ideg, NE);
  k_norm<<<(2 * N + TPB - 1) / TPB, TPB, 0, stream>>>(odeg, 2 * N);

  size_t nd  = (size_t)N * D;
  size_t nc  = (size_t)N * C;
  unsigned spmm_grd_d = (unsigned)(((size_t)NE * (D / 4) + TPB - 1) / TPB);
  unsigned spmm_grd_c = (unsigned)(((size_t)NE * (C / 4) + TPB - 1) / TPB);
  unsigned nd_grd = (unsigned)((nd + TPB - 1) / TPB);
  unsigned nc_grd = (unsigned)((nc + TPB - 1) / TPB);

  // --- layer 0 ---
  k_gemm_wmma<8, 128><<<gemm_grd, gemm_blk, 0, stream>>>(feat, odeg, W0, bufA, mtiles);
  k_zero<<<nd_grd, TPB, 0, stream>>>(bufB, nd);
  k_spmm<<<spmm_grd_d, TPB, 0, stream>>>(bufA, src, dst, bufB, NE, D);
  k_zero<<<1, TPB, 0, stream>>>(stats, 256);
  k_bn_stats<<<1024, D, 0, stream>>>(bufB, ideg, stats, N, D);
  k_bn_relu<128><<<nd_grd, TPB, 0, stream>>>(bufB, ideg, stats, g0, beta0, bufC, N);

  // --- layer 1 ---
  k_gemm_wmma<8, 128><<<gemm_grd, gemm_blk, 0, stream>>>(bufC, odeg, W1, bufA, mtiles);
  k_zero<<<nd_grd, TPB, 0, stream>>>(bufB, nd);
  k_spmm<<<spmm_grd_d, TPB, 0, stream>>>(bufA, src, dst, bufB, NE, D);
  k_zero<<<1, TPB, 0, stream>>>(stats, 256);
  k_bn_stats<<<1024, D, 0, stream>>>(bufB, ideg, stats, N, D);
  k_bn_relu<128><<<nd_grd, TPB, 0, stream>>>(bufB, ideg, stats, g1, beta1, bufC, N);

  // --- layer 2 (128 -> 40, no BN) ---
  k_gemm_wmma<3, 40><<<gemm_grd, gemm_blk, 0, stream>>>(bufC, odeg, W2, bufA, mtiles);
  k_zero<<<nc_grd, TPB, 0, stream>>>(bufB, nc);
  k_spmm<<<spmm_grd_c, TPB, 0, stream>>>(bufA, src, dst, bufB, NE, C);
  k_final<<<nc_grd, TPB, 0, stream>>>(bufB, ideg, b2, out, N, C);
}